// HierarchicalRewardFunction_58179626992080
// MI455X (gfx1250) — compile-verified
//
#include <hip/hip_runtime.h>
#include <math.h>

typedef __attribute__((ext_vector_type(2))) float v2f;
typedef __attribute__((ext_vector_type(8))) float v8f;

#define TPB 256
#define NWAVES (TPB / 32)
#define MAX_BLOCKS 2048

#define QUALITY_W 1.0f
#define UNCERT_W 0.5f
#define STEP_PEN 0.1f
#define EARLY_BONUS 0.5f
#define LATE_PEN 0.3f

__device__ __forceinline__ void process_elem(
    float cq, float pq, float u, int act, float sc, float fq,
    float late_thresh, float inv_ms, float alpha, float beta, float gamma,
    float* rs_o, float* rm_o, float* rl_o, float* tot_o, float acc[9]) {
  float rs = cq - pq;
  float stop_r = (u < 0.3f) ? EARLY_BONUS : ((u > 0.7f) ? -EARLY_BONUS : 0.0f);
  float cont_r = (u > 0.5f) ? (UNCERT_W * u) : (-UNCERT_W * (1.0f - u));
  cont_r -= (sc > late_thresh) ? LATE_PEN : 0.0f;
  float frac = sc * inv_ms;
  float rm = ((act == 0) ? stop_r : cont_r) - STEP_PEN * frac;
  float rl = QUALITY_W * fq + EARLY_BONUS * (1.0f - frac);
  *rs_o = rs; *rm_o = rm; *rl_o = rl;
  *tot_o = alpha * rs + beta * rm + gamma * rl;
  acc[0] += rs;    acc[1] += rm;    acc[2] += rl;
  acc[3] += rs*rs; acc[4] += rm*rm; acc[5] += rl*rl;
  acc[6] += rs*rm; acc[7] += rs*rl; acc[8] += rm*rl;
}

// Streaming map + per-block partial Gram sums.
// HBM-bound: 40 B/element => ~335 MB total => ~14.4 us floor at 23.3 TB/s.
__global__ void __launch_bounds__(TPB) reward_main_kernel(
    const float* __restrict__ cq, const float* __restrict__ pq,
    const float* __restrict__ u,  const int* __restrict__ act,
    const float* __restrict__ sc, const float* __restrict__ fq,
    const float* __restrict__ p_alpha, const float* __restrict__ p_beta,
    const float* __restrict__ p_gamma, const int* __restrict__ p_ms,
    int n,
    float* __restrict__ out_total, float* __restrict__ out_rs,
    float* __restrict__ out_rm,    float* __restrict__ out_rl,
    float* __restrict__ partials)  // MAX_BLOCKS x 16 floats
{
  const float alpha = *p_alpha, beta = *p_beta, gamma = *p_gamma;
  const float ms = (float)(*p_ms);
  const float inv_ms = 1.0f / ms;
  const float late_thresh = 0.8f * ms;

  float acc[9];
#pragma unroll
  for (int j = 0; j < 9; ++j) acc[j] = 0.0f;

  const int tid = threadIdx.x;
  const int gt = blockIdx.x * TPB + tid;
  const int stride = gridDim.x * TPB;
  const int n4 = n >> 2;

  // 128-bit vector loads (global_load_b128); total_reward stored as b128
  // (offset 0 is 16B aligned); r_short/r_mid/r_long live at offset n+1 etc.
  // (4B aligned only) => scalar b32 stores.
  for (int i = gt; i < n4; i += stride) {
    const float4 a = ((const float4*)cq)[i];
    const float4 b = ((const float4*)pq)[i];
    const float4 c = ((const float4*)u)[i];
    const int4   d = ((const int4*)act)[i];
    const float4 e = ((const float4*)sc)[i];
    const float4 f = ((const float4*)fq)[i];
    float rs0, rm0, rl0, rs1, rm1, rl1, rs2, rm2, rl2, rs3, rm3, rl3;
    float4 tot;
    process_elem(a.x, b.x, c.x, d.x, e.x, f.x, late_thresh, inv_ms, alpha, beta, gamma, &rs0, &rm0, &rl0, &tot.x, acc);
    process_elem(a.y, b.y, c.y, d.y, e.y, f.y, late_thresh, inv_ms, alpha, beta, gamma, &rs1, &rm1, &rl1, &tot.y, acc);
    process_elem(a.z, b.z, c.z, d.z, e.z, f.z, late_thresh, inv_ms, alpha, beta, gamma, &rs2, &rm2, &rl2, &tot.z, acc);
    process_elem(a.w, b.w, c.w, d.w, e.w, f.w, late_thresh, inv_ms, alpha, beta, gamma, &rs3, &rm3, &rl3, &tot.w, acc);
    ((float4*)out_total)[i] = tot;
    const int base = i << 2;
    out_rs[base + 0] = rs0; out_rs[base + 1] = rs1; out_rs[base + 2] = rs2; out_rs[base + 3] = rs3;
    out_rm[base + 0] = rm0; out_rm[base + 1] = rm1; out_rm[base + 2] = rm2; out_rm[base + 3] = rm3;
    out_rl[base + 0] = rl0; out_rl[base + 1] = rl1; out_rl[base + 2] = rl2; out_rl[base + 3] = rl3;
  }

  // Tail (n % 4), handled by block 0.
  const int tail = n & 3;
  if (blockIdx.x == 0 && tid < tail) {
    const int i = (n4 << 2) + tid;
    float rs, rm, rl, tot;
    process_elem(cq[i], pq[i], u[i], act[i], sc[i], fq[i],
                 late_thresh, inv_ms, alpha, beta, gamma, &rs, &rm, &rl, &tot, acc);
    out_total[i] = tot; out_rs[i] = rs; out_rm[i] = rm; out_rl[i] = rl;
  }

  // Deterministic reduction: wave32 shuffle tree, then LDS across 8 waves.
#pragma unroll
  for (int off = 16; off > 0; off >>= 1) {
#pragma unroll
    for (int j = 0; j < 9; ++j)
      acc[j] += __shfl_down(acc[j], off, 32);
  }

  __shared__ float red[NWAVES][9];
  const int wave = tid >> 5, lane = tid & 31;
  if (lane == 0) {
#pragma unroll
    for (int j = 0; j < 9; ++j) red[wave][j] = acc[j];
  }
  __syncthreads();
  if (tid < 16) {
    float s = 0.0f;
    if (tid < 9) {
#pragma unroll
      for (int w = 0; w < NWAVES; ++w) s += red[w][tid];
    }
    // Pad record to 16 floats with zeros so the WMMA finalize kernel can
    // load all 16 rows without touching poisoned workspace bytes.
    partials[blockIdx.x * 16 + tid] = (tid < 9) ? s : 0.0f;
  }
}

// Finalize: sum nblk per-block partial records (16 floats each, 9 valid) and
// compute ortho_loss. The column-sum is done with V_WMMA_F32_16X16X4_F32:
//   A (16x4, documented layout: lane m -> A[m][0..1] in the 2 A VGPRs,
//   lane m+16 -> A[m][2..3]) holds 4 block-partials per quantity (rows),
//   B = ones (layout independent), so D[m][n] = sum_k A[m][k].
// Accumulator chains across nblk/4 WMMA issues. D readout is documented:
// D[m][0] is VGPR m of lane 0 (m<8) and VGPR m-8 of lane 16 (m>=8).
__global__ void __launch_bounds__(32) reward_finalize_kernel(
    const float* __restrict__ partials, int nblk4, float fn,
    float* __restrict__ out_ortho)
{
  const int lane = threadIdx.x;      // 32 lanes, all active (EXEC all 1s)
  const int q = lane & 15;           // matrix row M = quantity index
  const int half = lane >> 4;        // 0 -> K=0,1 ; 1 -> K=2,3

  v8f c = {0.f, 0.f, 0.f, 0.f, 0.f, 0.f, 0.f, 0.f};
  v2f ones; ones.x = 1.0f; ones.y = 1.0f;

  for (int j = 0; j < nblk4; ++j) {
    const int b0 = 4 * j + 2 * half;
    v2f a;
    a.x = partials[(b0 + 0) * 16 + q];   // A[q][2*half]     -> A VGPR0
    a.y = partials[(b0 + 1) * 16 + q];   // A[q][2*half + 1] -> A VGPR1
    c = __builtin_amdgcn_wmma_f32_16x16x4_f32(
        /*neg_a=*/false, a, /*neg_b=*/false, ones,
        /*c_mod=*/(short)0, c, /*reuse_a=*/false, /*reuse_b=*/false);
  }

  float s[9];
#pragma unroll
  for (int r = 0; r < 8; ++r) s[r] = __shfl(c[r], 0, 32);  // D[0..7][0] @ lane 0
  s[8] = __shfl(c[0], 16, 32);                             // D[8][0]    @ lane 16

  if (lane == 0) {
    const float N = fn, Nm1 = fn - 1.0f;
    const float m1 = s[0] / N, m2 = s[1] / N, m3 = s[2] / N;
    const float sd1 = sqrtf(fmaxf((s[3] - N * m1 * m1) / Nm1, 0.0f)) + 1e-8f;
    const float sd2 = sqrtf(fmaxf((s[4] - N * m2 * m2) / Nm1, 0.0f)) + 1e-8f;
    const float sd3 = sqrtf(fmaxf((s[5] - N * m3 * m3) / Nm1, 0.0f)) + 1e-8f;
    const float c12 = (s[6] / N - m1 * m2) / (sd1 * sd2);
    const float c13 = (s[7] / N - m1 * m3) / (sd1 * sd3);
    const float c23 = (s[8] / N - m2 * m3) / (sd2 * sd3);
    *out_ortho = c12 * c12 + c13 * c13 + c23 * c23;
  }
}

extern "C" void kernel_launch(void* const* d_in, const int* in_sizes, int n_in,
                              void* d_out, int out_size, void* d_ws, size_t ws_size,
                              hipStream_t stream) {
  const float* cq  = (const float*)d_in[0];
  const float* pq  = (const float*)d_in[1];
  const float* u   = (const float*)d_in[2];
  const int*   act = (const int*)d_in[3];
  const float* sc  = (const float*)d_in[4];
  const float* fq  = (const float*)d_in[5];
  const float* pa  = (const float*)d_in[6];
  const float* pb  = (const float*)d_in[7];
  const float* pg  = (const float*)d_in[8];
  const int*   pms = (const int*)d_in[9];

  const int n = in_sizes[0];
  float* out = (float*)d_out;
  float* out_total = out;                       // [0, n)
  float* out_ortho = out + (long long)n;        // [n]
  float* out_rs    = out + (long long)n + 1;    // [n+1, 2n+1)
  float* out_rm    = out + 2LL * n + 1;         // [2n+1, 3n+1)
  float* out_rl    = out + 3LL * n + 1;         // [3n+1, 4n+1)

  // Fixed grid (clamped to workspace capacity, multiple of 4) => deterministic.
  int nblk = MAX_BLOCKS;
  int cap = (int)(ws_size / (16 * sizeof(float)));
  cap &= ~3;
  if (cap >= 4 && nblk > cap) nblk = cap;
  if (nblk < 4) nblk = 4;
  float* partials = (float*)d_ws;

  reward_main_kernel<<<nblk, TPB, 0, stream>>>(
      cq, pq, u, act, sc, fq, pa, pb, pg, pms, n,
      out_total, out_rs, out_rm, out_rl, partials);

  reward_finalize_kernel<<<1, 32, 0, stream>>>(partials, nblk / 4, (float)n,
                                               out_ortho);
}